// GATConvModel_7559142441204
// MI455X (gfx1250) — compile-verified
//
#include <hip/hip_runtime.h>
#include <cstdint>
#include <cstddef>

// ---------------------------------------------------------------------------
// GAT (3x GATConv + global_add_pool + FC) for MI455X / gfx1250, wave32.
// GEMMs use V_WMMA_F32_16X16X4_F32 (full fp32). Edge softmax/aggregation is
// L2-resident atomic scatter/gather (dominant cost; ~5 GB of L2 traffic).
// ---------------------------------------------------------------------------

typedef __attribute__((ext_vector_type(2))) float v2f;
typedef __attribute__((ext_vector_type(8))) float v8f;

#define DHID 256          // hidden width (fixed by the model)
#define NEG_SLOPE 0.2f

__device__ __forceinline__ v8f wmma_f32_k4(v2f a, v2f b, v8f c) {
  // D = A(16x4,f32) * B(4x16,f32) + C(16x16,f32)
  return __builtin_amdgcn_wmma_f32_16x16x4_f32(
      /*neg_a=*/false, a, /*neg_b=*/false, b,
      /*c_mod=*/(short)0, c, /*reuse_a=*/false, /*reuse_b=*/false);
}

// Order-preserving float -> uint key (for atomic max over signed floats)
__device__ __forceinline__ unsigned fkey(float f) {
  unsigned b = __float_as_uint(f);
  return (b & 0x80000000u) ? ~b : (b | 0x80000000u);
}
__device__ __forceinline__ float funkey(unsigned k) {
  unsigned b = (k & 0x80000000u) ? (k & 0x7FFFFFFFu) : ~k;
  return __uint_as_float(b);
}

// ---------------- H = X @ W  (X:[n,K] row-major, W:[K,256], H:[n,256]) ------
// 256 threads = 8 waves per block; block covers 16 rows x 256 cols.
// Each wave: 16x32 output tile (2 accumulators), K-loop in steps of 4.
__global__ __launch_bounds__(256) void gemm256(const float* __restrict__ X,
                                               const float* __restrict__ W,
                                               float* __restrict__ H,
                                               int n, int K) {
  const int wave = threadIdx.x >> 5;
  const int lane = threadIdx.x & 31;
  const int hi = lane >> 4;      // half-wave select
  const int lm = lane & 15;
  const int m0 = blockIdx.x * 16;
  const int n0 = wave * 32;

  int arow = m0 + lm;
  if (arow >= n) arow = n - 1;               // clamp (stores are guarded)
  const float* __restrict__ xp = X + (size_t)arow * K;

  v8f c0 = {}; v8f c1 = {};
  for (int k = 0; k < K; k += 4) {
    const int ka = k + 2 * hi;               // A/B ISA layout: K = vgpr + 2*hi
    v2f a;  a.x = xp[ka];                 a.y = xp[ka + 1];
    v2f b0; b0.x = W[(size_t)ka * DHID + n0 + lm];
            b0.y = W[(size_t)(ka + 1) * DHID + n0 + lm];
    v2f b1; b1.x = W[(size_t)ka * DHID + n0 + 16 + lm];
            b1.y = W[(size_t)(ka + 1) * DHID + n0 + 16 + lm];
    c0 = wmma_f32_k4(a, b0, c0);
    c1 = wmma_f32_k4(a, b1, c1);
  }
#pragma unroll
  for (int i = 0; i < 8; ++i) {
    const int m = m0 + i + 8 * hi;           // C/D layout: M = vgpr + 8*hi
    if (m < n) {
      H[(size_t)m * DHID + n0 + lm]      = c0[i];
      H[(size_t)m * DHID + n0 + 16 + lm] = c1[i];
    }
  }
}

// ---------------- s[n] = H[n].a_src ; d[n] = H[n].a_dst --------------------
// One wave per node; 8 cols per lane (float4 x2); shfl-xor reduction.
__global__ __launch_bounds__(256) void rowdot(const float* __restrict__ H,
                                              const float* __restrict__ asrc,
                                              const float* __restrict__ adst,
                                              float* __restrict__ sv,
                                              float* __restrict__ dv, int n) {
  const int gid  = blockIdx.x * blockDim.x + threadIdx.x;
  const int node = gid >> 5;
  const int lane = gid & 31;
  if (node >= n) return;
  const float4* __restrict__ hr = (const float4*)(H + (size_t)node * DHID);
  const float4* __restrict__ as = (const float4*)asrc;
  const float4* __restrict__ ad = (const float4*)adst;
  float4 h0 = hr[lane * 2], h1 = hr[lane * 2 + 1];
  float4 s0 = as[lane * 2], s1 = as[lane * 2 + 1];
  float4 d0 = ad[lane * 2], d1 = ad[lane * 2 + 1];
  float s = h0.x*s0.x + h0.y*s0.y + h0.z*s0.z + h0.w*s0.w
          + h1.x*s1.x + h1.y*s1.y + h1.z*s1.z + h1.w*s1.w;
  float d = h0.x*d0.x + h0.y*d0.y + h0.z*d0.z + h0.w*d0.w
          + h1.x*d1.x + h1.y*d1.y + h1.z*d1.z + h1.w*d1.w;
#pragma unroll
  for (int off = 16; off > 0; off >>= 1) {
    s += __shfl_xor(s, off, 32);
    d += __shfl_xor(d, off, 32);
  }
  if (lane == 0) { sv[node] = s; dv[node] = d; }
}

// ---------------- segment max of leaky-relu edge logits --------------------
__global__ __launch_bounds__(256) void edge_max(const int* __restrict__ src,
                                                const int* __restrict__ dst,
                                                const float* __restrict__ sv,
                                                const float* __restrict__ dv,
                                                unsigned* __restrict__ mkey,
                                                int E, int total) {
  const int i = blockIdx.x * blockDim.x + threadIdx.x;
  if (i >= total) return;
  const int si = (i < E) ? src[i] : (i - E);   // appended self-loops
  const int di = (i < E) ? dst[i] : (i - E);
  float e = sv[si] + dv[di];
  e = (e >= 0.f) ? e : NEG_SLOPE * e;
  atomicMax(&mkey[di], fkey(e));
}

// ---------------- ex = exp(e - m[dst]); den[dst] += ex ---------------------
__global__ __launch_bounds__(256) void edge_expsum(const int* __restrict__ src,
                                                   const int* __restrict__ dst,
                                                   const float* __restrict__ sv,
                                                   const float* __restrict__ dv,
                                                   const unsigned* __restrict__ mkey,
                                                   float* __restrict__ ex,
                                                   float* __restrict__ den,
                                                   int E, int total) {
  const int i = blockIdx.x * blockDim.x + threadIdx.x;
  if (i >= total) return;
  const int si = (i < E) ? src[i] : (i - E);
  const int di = (i < E) ? dst[i] : (i - E);
  float e = sv[si] + dv[di];
  e = (e >= 0.f) ? e : NEG_SLOPE * e;
  const float v = __expf(e - funkey(mkey[di]));
  ex[i] = v;
  atomicAdd(&den[di], v);
}

// ---------------- agg[dst] += (ex/den[dst]) * H[src] -----------------------
// One wave per edge; 8 floats/lane: 2 float4 gathers + 8 f32 atomics (L2-hot).
__global__ __launch_bounds__(256) void edge_agg(const int* __restrict__ src,
                                                const int* __restrict__ dst,
                                                const float* __restrict__ ex,
                                                const float* __restrict__ den,
                                                const float* __restrict__ H,
                                                float* __restrict__ agg,
                                                int E, int total) {
  const int gid = blockIdx.x * blockDim.x + threadIdx.x;
  const int w = gid >> 5;
  const int lane = gid & 31;
  if (w >= total) return;
  const int si = (w < E) ? src[w] : (w - E);
  const int di = (w < E) ? dst[w] : (w - E);
  const float alpha = ex[w] / den[di];
  const float4* __restrict__ hs = (const float4*)(H + (size_t)si * DHID);
  float4 a0 = hs[lane * 2], a1 = hs[lane * 2 + 1];
  float* ap = agg + (size_t)di * DHID + lane * 8;
  atomicAdd(ap + 0, alpha * a0.x); atomicAdd(ap + 1, alpha * a0.y);
  atomicAdd(ap + 2, alpha * a0.z); atomicAdd(ap + 3, alpha * a0.w);
  atomicAdd(ap + 4, alpha * a1.x); atomicAdd(ap + 5, alpha * a1.y);
  atomicAdd(ap + 6, alpha * a1.z); atomicAdd(ap + 7, alpha * a1.w);
}

// ---------------- X = relu(X + b) (in place) -------------------------------
__global__ __launch_bounds__(256) void bias_relu(float* __restrict__ X,
                                                 const float* __restrict__ b,
                                                 long total) {
  const long i = (long)blockIdx.x * blockDim.x + threadIdx.x;
  if (i >= total) return;
  const float v = X[i] + b[i & (DHID - 1)];
  X[i] = (v > 0.f) ? v : 0.f;
}

// ---------------- pooled[batch[n]] += X[n] ---------------------------------
__global__ __launch_bounds__(256) void pool_sum(const float* __restrict__ X,
                                                const int* __restrict__ batch,
                                                float* __restrict__ pooled,
                                                long total) {
  const long i = (long)blockIdx.x * blockDim.x + threadIdx.x;
  if (i >= total) return;
  const int node = (int)(i >> 8);
  const int c = (int)(i & (DHID - 1));
  atomicAdd(&pooled[(size_t)batch[node] * DHID + c], X[i]);
}

// ---------------- out = pooled @ fcW + fcb  (64x256 @ 256x64) --------------
// 512 threads = 16 waves; wave (mt,nt) owns one 16x16 tile; K=256 in steps of 4.
__global__ __launch_bounds__(512) void final_gemm(const float* __restrict__ P,
                                                  const float* __restrict__ Wf,
                                                  const float* __restrict__ bf,
                                                  float* __restrict__ out,
                                                  int G, int DOUT) {
  const int wave = threadIdx.x >> 5;
  const int lane = threadIdx.x & 31;
  const int hi = lane >> 4, lm = lane & 15;
  const int mt = wave >> 2, nt = wave & 3;
  if (mt * 16 >= G || nt * 16 >= DOUT) return;
  v8f c = {};
  const float* __restrict__ pr = P + (size_t)(mt * 16 + lm) * DHID;
  for (int k = 0; k < DHID; k += 4) {
    const int ka = k + 2 * hi;
    v2f a; a.x = pr[ka]; a.y = pr[ka + 1];
    v2f b; b.x = Wf[(size_t)ka * DOUT + nt * 16 + lm];
           b.y = Wf[(size_t)(ka + 1) * DOUT + nt * 16 + lm];
    c = wmma_f32_k4(a, b, c);
  }
#pragma unroll
  for (int i = 0; i < 8; ++i) {
    const int m = mt * 16 + i + 8 * hi;
    const int nn = nt * 16 + lm;
    if (m < G && nn < DOUT) out[(size_t)m * DOUT + nn] = c[i] + bf[nn];
  }
}

// ===========================================================================
extern "C" void kernel_launch(void* const* d_in, const int* in_sizes, int n_in,
                              void* d_out, int out_size, void* d_ws, size_t ws_size,
                              hipStream_t stream) {
  const float* x    = (const float*)d_in[0];
  const int* eidx   = (const int*)d_in[1];
  const int* batch  = (const int*)d_in[2];
  const float* Wl[3]    = {(const float*)d_in[3], (const float*)d_in[7],  (const float*)d_in[11]};
  const float* aSrc[3]  = {(const float*)d_in[4], (const float*)d_in[8],  (const float*)d_in[12]};
  const float* aDst[3]  = {(const float*)d_in[5], (const float*)d_in[9],  (const float*)d_in[13]};
  const float* bl[3]    = {(const float*)d_in[6], (const float*)d_in[10], (const float*)d_in[14]};
  const float* fcW  = (const float*)d_in[15];
  const float* fcb  = (const float*)d_in[16];
  float* out = (float*)d_out;

  const int N   = in_sizes[2];            // batch vector length = #nodes
  const int DIN = in_sizes[0] / N;        // 128
  const int E   = in_sizes[1] / 2;        // 800000
  const int DOUT = in_sizes[16];          // 64
  const int G    = out_size / DOUT;       // 64
  const int TOT  = E + N;                 // edges incl. self-loops
  const int* eSrc = eidx;                 // edge_index[0]
  const int* eDst = eidx + E;             // edge_index[1]

  // ---- workspace carve-out (all offsets keep 16B alignment) ----
  float* H   = (float*)d_ws;                  // [N,256] layer GEMM output
  float* XB  = H + (size_t)N * DHID;          // [N,256] agg / next-layer input
  float* sv  = XB + (size_t)N * DHID;         // [N]
  float* dv  = sv + N;                        // [N]
  unsigned* mkey = (unsigned*)(dv + N);       // [N]
  float* den = (float*)(mkey + N);            // [N]
  float* ex  = den + N;                       // [E+N]
  float* pooled = ex + TOT;                   // [G,256]

  const int mblocks    = (N + 15) / 16;                 // GEMM row stripes
  const int rowBlocks  = (int)(((long)N * 32 + 255) / 256);
  const int edgeBlocks = (TOT + 255) / 256;
  const int aggBlocks  = (int)(((long)TOT * 32 + 255) / 256);
  const long elems     = (long)N * DHID;
  const int elemBlocks = (int)((elems + 255) / 256);

  const float* Xin = x;   // layer-1 input
  int Kin = DIN;

  for (int L = 0; L < 3; ++L) {
    gemm256<<<mblocks, 256, 0, stream>>>(Xin, Wl[L], H, N, Kin);
    rowdot<<<rowBlocks, 256, 0, stream>>>(H, aSrc[L], aDst[L], sv, dv, N);
    hipMemsetAsync(mkey, 0, (size_t)N * sizeof(unsigned), stream);
    edge_max<<<edgeBlocks, 256, 0, stream>>>(eSrc, eDst, sv, dv, mkey, E, TOT);
    hipMemsetAsync(den, 0, (size_t)N * sizeof(float), stream);
    edge_expsum<<<edgeBlocks, 256, 0, stream>>>(eSrc, eDst, sv, dv, mkey, ex, den, E, TOT);
    hipMemsetAsync(XB, 0, (size_t)N * DHID * sizeof(float), stream);
    edge_agg<<<aggBlocks, 256, 0, stream>>>(eSrc, eDst, ex, den, H, XB, E, TOT);
    bias_relu<<<elemBlocks, 256, 0, stream>>>(XB, bl[L], elems);
    Xin = XB;           // next layer reads the aggregated+activated features
    Kin = DHID;
  }

  hipMemsetAsync(pooled, 0, (size_t)G * DHID * sizeof(float), stream);
  pool_sum<<<elemBlocks, 256, 0, stream>>>(XB, batch, pooled, elems);
  final_gemm<<<1, 512, 0, stream>>>(pooled, fcW, fcb, out, G, DOUT);
}